// Eff_TTEmbedding_72825465471567
// MI455X (gfx1250) — compile-verified
//
#include <hip/hip_runtime.h>
#include <stdint.h>

typedef __attribute__((ext_vector_type(2))) float v2f;
typedef __attribute__((ext_vector_type(4))) float v4f;
typedef __attribute__((ext_vector_type(8))) float v8f;

#define WAVES_PER_BLOCK 8

// One wave32 per sample. fp32 WMMA (V_WMMA_F32_16X16X4_F32) for both TT contractions.
__global__ __launch_bounds__(256) void tt_emb_wmma_kernel(
    const int64_t* __restrict__ indices,
    const float* __restrict__ core0,   // (100, 128)  -> g0 (4x32)
    const float* __restrict__ core1,   // (100, 4096) -> g1 (32x128)
    const float* __restrict__ core2,   // (100, 256)  -> g2 (32x8)
    float* __restrict__ out,           // (B, 128)
    int batch)
{
    __shared__ float lds_t[WAVES_PER_BLOCK * 512];

    const int lane = threadIdx.x & 31;
    const int wave = threadIdx.x >> 5;
    const int s = blockIdx.x * WAVES_PER_BLOCK + wave;
    if (s >= batch) return;                 // wave-uniform

    const int nloc = lane & 15;             // M or N position within 16
    const int kh   = (lane >> 4) << 1;      // k sub-offset: 0 (lanes 0-15), 2 (lanes 16-31)

    // Mixed-radix index decomposition over P=[100,100,100]
    const int idx = (int)indices[s];
    const int i0  = idx / 10000;
    const int rem = idx - i0 * 10000;
    const int i1  = rem / 100;
    const int i2  = rem - i1 * 100;

    const float* g0 = core0 + i0 * 128;     // row-major (4 x 32)
    const float* g1 = core1 + i1 * 4096;    // row-major (32 x 128)
    const float* g2 = core2 + i2 * 256;     // row-major (32 x 8)

    float* tbuf = &lds_t[wave * 512];

    // ---- Preload A fragments for GEMM1 (g0: 4 valid rows, zero-pad M=4..15) ----
    // 16x4 fp32 A layout: lane L holds M = L%16, k = kk*4 + kh + v (v = vgpr 0/1)
    v2f afrag[8];
#pragma unroll
    for (int kk = 0; kk < 8; ++kk) {
        v2f a = {0.0f, 0.0f};
        if (nloc < 4) {
            const float* p = g0 + nloc * 32 + kk * 4 + kh;
            a.x = p[0];
            a.y = p[1];
        }
        afrag[kk] = a;
    }

    // ---- GEMM1: t(4x128) = g0(4x32) x g1(32x128); 8 N-tiles x 8 K-steps ----
#pragma unroll
    for (int nn = 0; nn < 8; ++nn) {
        v8f acc = {};
#pragma unroll
        for (int kk = 0; kk < 8; ++kk) {
            // 4x16 fp32 B layout: lane L holds N = L%16, k = kk*4 + kh + v
            const float* p = g1 + (kk * 4 + kh) * 128 + nn * 16 + nloc;
            v2f b;
            b.x = p[0];
            b.y = p[128];
            acc = __builtin_amdgcn_wmma_f32_16x16x4_f32(
                false, afrag[kk], false, b, (short)0, acc, false, false);
        }
        // D rows M=0..3 live in acc[0..3], lanes 0-15 (N = lane). Stage t into LDS.
        if (lane < 16) {
            tbuf[0 * 128 + nn * 16 + lane] = acc[0];
            tbuf[1 * 128 + nn * 16 + lane] = acc[1];
            tbuf[2 * 128 + nn * 16 + lane] = acc[2];
            tbuf[3 * 128 + nn * 16 + lane] = acc[3];
        }
    }

    // ---- GEMM2: out(16x8) = t(16x32) x g2(32x8); N zero-padded to 16 ----
    // A row M = q0*4+q1 -> t element at tbuf[(M>>2)*128 + (M&3)*32 + r2]
    v8f dacc = {};
    const float* arow = tbuf + (nloc >> 2) * 128 + (nloc & 3) * 32;
#pragma unroll
    for (int kk = 0; kk < 8; ++kk) {
        v2f a;                                    // consecutive pair -> ds b64 load
        a.x = arow[kk * 4 + kh];
        a.y = arow[kk * 4 + kh + 1];
        v2f b = {0.0f, 0.0f};
        if (nloc < 8) {                           // N=8..15 are pad columns
            const float* p = g2 + (kk * 4 + kh) * 8 + nloc;
            b.x = p[0];
            b.y = p[8];
        }
        dacc = __builtin_amdgcn_wmma_f32_16x16x4_f32(
            false, a, false, b, (short)0, dacc, false, false);
    }

    // ---- Stage 16x8 result through LDS; emit one coalesced b128 store per lane ----
    if (nloc < 8) {
        const int mhi = (lane >> 4) << 3;         // +8 for upper lane half
#pragma unroll
        for (int j = 0; j < 8; ++j) {
            tbuf[(j + mhi) * 8 + nloc] = dacc[j]; // out flat = M*8 + q2
        }
    }
    // Same-wave LDS ops are in-order; compiler inserts s_wait_dscnt for the reload.
    v4f r = *(const v4f*)&tbuf[lane * 4];
    *(v4f*)(out + (size_t)s * 128 + lane * 4) = r;
}

extern "C" void kernel_launch(void* const* d_in, const int* in_sizes, int n_in,
                              void* d_out, int out_size, void* d_ws, size_t ws_size,
                              hipStream_t stream) {
    const int64_t* indices = (const int64_t*)d_in[0];
    const float*   core0   = (const float*)d_in[1];
    const float*   core1   = (const float*)d_in[2];
    const float*   core2   = (const float*)d_in[3];
    float*         out     = (float*)d_out;

    const int batch  = in_sizes[0];
    const int blocks = (batch + WAVES_PER_BLOCK - 1) / WAVES_PER_BLOCK;
    tt_emb_wmma_kernel<<<blocks, 32 * WAVES_PER_BLOCK, 0, stream>>>(
        indices, core0, core1, core2, out, batch);
}